// OutputLayer_55697135894703
// MI455X (gfx1250) — compile-verified
//
#include <hip/hip_runtime.h>
#include <stdint.h>

// Problem constants (match the reference)
#define A_N 200000
#define C_N 80
#define K_N 300
#define KP  304            // K padded to 19 WMMA tiles of 16
#define NTILE 19
#define NT2 1024           // threads: top-k kernel (32 waves)
#define NT3 512            // threads: nms kernel (16 waves)
#define CAND_MAX 1024
#define SCORE_TH 0.1f

typedef unsigned int u32;
typedef unsigned long long u64;
typedef float v2f __attribute__((ext_vector_type(2)));
typedef float v8f __attribute__((ext_vector_type(8)));
typedef unsigned int u32x4 __attribute__((ext_vector_type(4)));
typedef int i32x4 __attribute__((ext_vector_type(4)));
typedef int i32x8 __attribute__((ext_vector_type(8)));

#if defined(__gfx1250__) && __has_builtin(__builtin_amdgcn_tensor_load_to_lds)
#define HAVE_TDM 1
#else
#define HAVE_TDM 0
#endif

// ---------------------------------------------------------------------------
// CDNA5 async global->LDS (per-lane gather, ASYNCcnt-tracked)
// global_load_async_to_lds_b32 vdst(lds byte addr), vaddr(u32 off), s[base]
// ---------------------------------------------------------------------------
__device__ __forceinline__ void async_ld_b32(u64 sbase, u32 voff, u32 ldsoff) {
  asm volatile("global_load_async_to_lds_b32 %0, %1, %2"
               :: "v"(ldsoff), "v"(voff), "s"(sbase)
               : "memory");
}

// Double-buffered async-staged scan over one class's strided score column.
// Each lane stages its own LDS slot; no cross-lane sharing => no barriers.
template <typename F>
__device__ __forceinline__ void scan_class(u64 base, int tid, float* stage, F fn) {
  const int nch = (A_N + NT2 - 1) / NT2;  // 196 chunks
  const u32 lds0 = (u32)(uintptr_t)(stage + tid);
  const u32 lds1 = (u32)(uintptr_t)(stage + NT2 + tid);
  auto issue = [&](int ch) {
    // retire any LDS read of the buffer we are about to overwrite
    asm volatile("s_wait_dscnt 0" ::: "memory");
    int a = ch * NT2 + tid;
    if (a > A_N - 1) a = A_N - 1;          // clamp (keeps EXEC full)
    async_ld_b32(base, (u32)a * (u32)(C_N * 4), (ch & 1) ? lds1 : lds0);
  };
  issue(0);
  for (int ch = 0; ch < nch; ++ch) {
    if (ch + 1 < nch) {
      issue(ch + 1);                        // prefetch next buffer
      asm volatile("s_wait_asynccnt 1" ::: "memory");  // chunk ch complete
    } else {
      asm volatile("s_wait_asynccnt 0" ::: "memory");
    }
    const float v = stage[(ch & 1) * NT2 + tid];
    const int a = ch * NT2 + tid;
    if (a < A_N) fn(a, v);
  }
}

#if HAVE_TDM
// ---------------------------------------------------------------------------
// TDM-staged scan: one tensor_load_to_lds per 1024-element chunk DMAs the
// strided column (tile 1 x 1024, tensor_dim0_stride = C_N elements) into
// contiguous LDS. Tail chunk zero-filled by TDM OOB rule via tensor_dim1.
// D# packed per ISA 8.3/8.4: group0 {count=1, lds_addr, global_addr, type=2},
// group1 {data_size=4B, tensor_dim0=1, tensor_dim1=rows_left, tile_dim0=1,
//         tile_dim1=NT2, tensor_dim0_stride=C_N}.
// This toolchain's builtin is the 6-arg form:
//   (u32x4 g0, i32x8 g1, i32x4 g2, i32x4 g3, i32x8, i32 cpol)
// ---------------------------------------------------------------------------
__device__ __forceinline__ void tdm_load_column(u32 lds_off, u64 gaddr,
                                                u32 rows_left, u32 rows_tile) {
  u32x4 g0;
  g0[0] = 1u;                                            // count=1 (user D#)
  g0[1] = lds_off;                                       // lds_addr (bytes)
  g0[2] = (u32)(gaddr & 0xFFFFFFFFull);                  // global_addr[31:0]
  g0[3] = (u32)((gaddr >> 32) & 0x1FFFFFFull)            // global_addr[56:32]
          | (2u << 30);                                  // type=2 ("image")
  i32x8 g1;
  g1[0] = (int)(2u << 16);                               // data_size=2 -> 4B
  g1[1] = (int)(1u << 16);                               // tensor_dim0 = 1
  g1[2] = (int)((rows_left & 0xFFFFu) << 16);            // tensor_dim1 lo16
  g1[3] = (int)(((rows_left >> 16) & 0xFFFFu)            // tensor_dim1 hi16
          | (1u << 16));                                 // tile_dim0 = 1
  g1[4] = (int)(rows_tile & 0xFFFFu);                    // tile_dim1 (tile_dim2=0)
  g1[5] = (int)(u32)C_N;                                 // dim0 stride lo32 = 80
  g1[6] = 0;                                             // stride hi | dim1stride lo
  g1[7] = 0;                                             // dim1 stride hi
  i32x4 z4 = {0, 0, 0, 0};
  i32x8 z8 = {0, 0, 0, 0, 0, 0, 0, 0};
  __builtin_amdgcn_tensor_load_to_lds(g0, g1, z4, z4, z8, 0);
}

template <typename F>
__device__ __forceinline__ void scan_class_tdm(u64 base, int tid, float* stage,
                                               F fn) {
  const int nch = (A_N + NT2 - 1) / NT2;
  const u32 lds0 = (u32)(uintptr_t)stage;
  for (int ch = 0; ch < nch; ++ch) {
    if (tid < 32) {  // wave 0 drives the DMA (TDM ignores EXEC; 1 op/wave)
      const u32 row0 = (u32)ch * NT2;
      tdm_load_column(lds0, base + (u64)row0 * (u32)(C_N * 4),
                      (u32)A_N - row0, NT2);
      __builtin_amdgcn_s_wait_tensorcnt(0);
    }
    __syncthreads();                       // publish staged chunk to all waves
    const int a = ch * NT2 + tid;
    if (a < A_N) fn(a, stage[tid]);
    __syncthreads();                       // chunk consumed; safe to overwrite
  }
}
#endif  // HAVE_TDM

// ---------------------------------------------------------------------------
// Kernel 1: per-class exact top-K (two-level radix select + bitonic sort),
// then decode boxes only for the K survivors (saves decoding 200k boxes).
// One block per class.
// ---------------------------------------------------------------------------
__global__ __launch_bounds__(NT2) void retina_topk_decode(
    const float* __restrict__ cls, const float* __restrict__ reg,
    const float* __restrict__ anc, float* __restrict__ tk_score,
    float* __restrict__ tk_box) {
  __shared__ u32 hist[4096];
  __shared__ u64 skey[CAND_MAX];
  __shared__ float stage[2 * NT2];
  __shared__ int info[4];  // [0]=bucket1 [1]=countAbove1 [2]=thr_lo [3]=cand cnt

  const int tid = threadIdx.x;
  const int c = blockIdx.x;
  const u64 base = (u64)(uintptr_t)(cls + c);  // class column base

  // -- pass 1: histogram on float bits [30:19] (monotone for scores >= 0)
  for (int i = tid; i < 4096; i += NT2) hist[i] = 0u;
  __syncthreads();
#if HAVE_TDM
  scan_class_tdm(base, tid, stage, [&](int a, float v) {
#else
  scan_class(base, tid, stage, [&](int a, float v) {
#endif
    (void)a;
    u32 u = __float_as_uint(v);
    atomicAdd(&hist[(u >> 19) & 4095u], 1u);
  });
  __syncthreads();
  if (tid == 0) {
    u32 cum = 0; int b = 4095;
    for (; b > 0; --b) { u32 h = hist[b]; if (cum + h >= (u32)K_N) break; cum += h; }
    info[0] = b; info[1] = (int)cum;
  }
  __syncthreads();
  const u32 b1 = (u32)info[0];
  const u32 above1 = (u32)info[1];
  __syncthreads();

  // -- pass 2: refine inside boundary bucket on bits [18:7]
  for (int i = tid; i < 4096; i += NT2) hist[i] = 0u;
  __syncthreads();
  scan_class(base, tid, stage, [&](int a, float v) {
    (void)a;
    u32 u = __float_as_uint(v);
    if ((u >> 19) == b1) atomicAdd(&hist[(u >> 7) & 4095u], 1u);
  });
  __syncthreads();
  if (tid == 0) {
    u32 cum = above1; int b = 4095;
    for (; b > 0; --b) { u32 h = hist[b]; if (cum + h >= (u32)K_N) break; cum += h; }
    info[2] = (int)((b1 << 19) | ((u32)b << 7));
    info[3] = 0;
  }
  __syncthreads();
  const u32 thr_lo = (u32)info[2];

  // -- pass 3: collect candidates (>= threshold), key packs score + ~idx so a
  //    single descending sort reproduces lax.top_k tie order (low idx first).
  scan_class(base, tid, stage, [&](int a, float v) {
    u32 u = __float_as_uint(v);
    if (u >= thr_lo) {
      int p = atomicAdd(&info[3], 1);
      if (p < CAND_MAX) skey[p] = ((u64)u << 32) | (u64)(u32)(~(u32)a);
    }
  });
  __syncthreads();
  {
    int cnt = info[3]; if (cnt > CAND_MAX) cnt = CAND_MAX;
    for (int i = tid; i < CAND_MAX; i += NT2)
      if (i >= cnt) skey[i] = 0ull;
  }
  __syncthreads();

  // -- bitonic sort, descending, 1024 keys, 1024 threads
  for (u32 k = 2; k <= CAND_MAX; k <<= 1) {
    for (u32 j = k >> 1; j > 0; j >>= 1) {
      const u32 i = (u32)tid, ixj = i ^ j;
      if (ixj > i) {
        const u64 x = skey[i], y = skey[ixj];
        const bool up = ((i & k) == 0u);
        if (up ? (x < y) : (x > y)) { skey[i] = y; skey[ixj] = x; }
      }
      __syncthreads();
    }
  }

  // -- emit top-K with on-the-fly box decode (RetinaNet BBoxTransform)
  if (tid < K_N) {
    const u64 key = skey[tid];
    const float sc = __uint_as_float((u32)(key >> 32));
    u32 idx = (key == 0ull) ? 0u : (u32)(~(u32)key);
    if (idx >= (u32)A_N) idx = 0u;
    const float ax1 = anc[idx * 4 + 0], ay1 = anc[idx * 4 + 1];
    const float ax2 = anc[idx * 4 + 2], ay2 = anc[idx * 4 + 3];
    const float dx = reg[idx * 4 + 0] * 0.1f, dy = reg[idx * 4 + 1] * 0.1f;
    const float dw = reg[idx * 4 + 2] * 0.2f, dh = reg[idx * 4 + 3] * 0.2f;
    const float w = ax2 - ax1, h = ay2 - ay1;
    const float pcx = ax1 + 0.5f * w + dx * w;
    const float pcy = ay1 + 0.5f * h + dy * h;
    const float pw = __expf(dw) * w, ph = __expf(dh) * h;
    const int o = c * K_N + tid;
    tk_score[o] = sc;
    tk_box[o * 4 + 0] = pcx - 0.5f * pw;
    tk_box[o * 4 + 1] = pcy - 0.5f * ph;
    tk_box[o * 4 + 2] = pcx + 0.5f * pw;
    tk_box[o * 4 + 3] = pcy + 0.5f * ph;
  }
}

// ---------------------------------------------------------------------------
// Kernel 2: per-class NMS + final outputs. One block per class.
// 16x16 IoU tiles: union denominator area_i+area_j computed by one
// v_wmma_f32_16x16x4_f32 (outer-sum via A=[area,1], B=[1;area]); intersection
// via VALU; "iou > thr" done division-free as inter > thr*union (union > 0);
// bits packed with ballot; greedy suppression as 1-wave serial scan.
// ---------------------------------------------------------------------------
__global__ __launch_bounds__(NT3) void retina_nms(
    const float* __restrict__ tk_score, const float* __restrict__ tk_box,
    const float* __restrict__ nms_thr_p, float* __restrict__ out_scores,
    float* __restrict__ out_classes, float* __restrict__ out_boxes) {
  __shared__ float sb[KP * 4];
  __shared__ float sarea[KP];
  __shared__ float sscore[KP];
  __shared__ u32 rowmask[KP * NTILE];  // 16 col-bits per (row, col-tile)
  __shared__ u32 ssupp[NTILE];

  const int tid = threadIdx.x;
  const int c = blockIdx.x;
  const int wave = tid >> 5, lane = tid & 31;
  const float thr = nms_thr_p[0];

  for (int t = tid; t < KP; t += NT3) {
    float x1 = 0.f, y1 = 0.f, x2 = 0.f, y2 = 0.f, s = 0.f;
    if (t < K_N) {
      const int o = c * K_N + t;
      s = tk_score[o];
      x1 = tk_box[o * 4 + 0]; y1 = tk_box[o * 4 + 1];
      x2 = tk_box[o * 4 + 2]; y2 = tk_box[o * 4 + 3];
    }
    sb[t * 4 + 0] = x1; sb[t * 4 + 1] = y1;
    sb[t * 4 + 2] = x2; sb[t * 4 + 3] = y2;
    sscore[t] = s;
    sarea[t] = (x2 - x1) * (y2 - y1);
  }
  for (int i = tid; i < KP * NTILE; i += NT3) rowmask[i] = 0u;
  __syncthreads();

  // upper-triangular 16x16 tiles; loop bounds wave-uniform => EXEC full at WMMA
  for (int t = wave; t < NTILE * NTILE; t += NT3 / 32) {
    const int ti = t / NTILE, tj = t % NTILE;
    if (tj < ti) continue;  // wave-uniform skip (strictly-later cols only)
    const int rbase = ti * 16, cbase = tj * 16;

    v2f av, bv;
    if (lane < 16) {
      av.x = sarea[rbase + lane]; av.y = 1.0f;   // A[i][0]=area_i, A[i][1]=1
      bv.x = 1.0f; bv.y = sarea[cbase + lane];   // B[0][j]=1,      B[1][j]=area_j
    } else {
      av.x = 0.f; av.y = 0.f; bv.x = 0.f; bv.y = 0.f;  // K=2,3 lanes zero
    }
    v8f acc = {};
    acc = __builtin_amdgcn_wmma_f32_16x16x4_f32(
        false, av, false, bv, (short)0, acc, false, false);
    // acc[r] = area_row(rbase+r+8*(lane>>4)) + area_col(cbase+(lane&15))

    const int col = cbase + (lane & 15);
    const float cx1 = sb[col * 4 + 0], cy1 = sb[col * 4 + 1];
    const float cx2 = sb[col * 4 + 2], cy2 = sb[col * 4 + 3];
    const int rhalf = (lane >> 4) * 8;
#pragma unroll
    for (int r = 0; r < 8; ++r) {
      const int row = rbase + rhalf + r;
      const float rx1 = sb[row * 4 + 0], ry1 = sb[row * 4 + 1];
      const float rx2 = sb[row * 4 + 2], ry2 = sb[row * 4 + 3];
      const float iw = fmaxf(fminf(rx2, cx2) - fmaxf(rx1, cx1), 0.f);
      const float ih = fmaxf(fminf(ry2, cy2) - fmaxf(ry1, cy1), 0.f);
      const float inter = iw * ih;
      const float uni = acc[r] - inter + 1e-9f;       // union > 0 always
      const bool pred = (inter > thr * uni) &&        // iou > thr, no divide
                        (col > row) && (col < K_N) && (row < K_N);
      const u32 bal = __builtin_amdgcn_ballot_w32(pred);
      if (lane == 0) {
        rowmask[(rbase + r) * NTILE + tj] = bal & 0xFFFFu;       // rows 0..7
        rowmask[(rbase + r + 8) * NTILE + tj] = bal >> 16;       // rows 8..15
      }
    }
  }
  __syncthreads();

  // greedy suppression: wave 0; lane w owns 16 suppression bits (cols w*16..)
  if (tid < 32) {
    u32 supp = 0u;
    for (int i = 0; i < K_N; ++i) {
      const u32 w = (u32)__builtin_amdgcn_ds_bpermute((i >> 4) * 4, (int)supp);
      const bool keep = (sscore[i] > SCORE_TH) && !((w >> (i & 15)) & 1u);
      if (keep && lane < NTILE) supp |= rowmask[i * NTILE + lane];
    }
    if (lane < NTILE) ssupp[lane] = supp;
  }
  __syncthreads();

  if (tid < K_N) {
    const bool sup = (ssupp[tid >> 4] >> (tid & 15)) & 1u;
    const bool keep = (sscore[tid] > SCORE_TH) && !sup;
    const int o = c * K_N + tid;
    out_scores[o] = keep ? sscore[tid] : 0.f;
    out_classes[o] = keep ? (float)c : -1.f;
    out_boxes[o * 4 + 0] = keep ? sb[tid * 4 + 0] : 0.f;
    out_boxes[o * 4 + 1] = keep ? sb[tid * 4 + 1] : 0.f;
    out_boxes[o * 4 + 2] = keep ? sb[tid * 4 + 2] : 0.f;
    out_boxes[o * 4 + 3] = keep ? sb[tid * 4 + 3] : 0.f;
  }
}

// ---------------------------------------------------------------------------
extern "C" void kernel_launch(void* const* d_in, const int* in_sizes, int n_in,
                              void* d_out, int out_size, void* d_ws,
                              size_t ws_size, hipStream_t stream) {
  (void)in_sizes; (void)n_in; (void)out_size; (void)ws_size;
  const float* cls = (const float*)d_in[0];  // [1,A,C]
  const float* reg = (const float*)d_in[1];  // [1,A,4]
  const float* anc = (const float*)d_in[2];  // [1,A,4]
  const float* nms = (const float*)d_in[3];  // scalar

  float* ws = (float*)d_ws;
  float* tk_score = ws;                       // C*K
  float* tk_box = ws + C_N * K_N;             // C*K*4

  float* out_scores = (float*)d_out;                    // C*K
  float* out_classes = out_scores + C_N * K_N;          // C*K
  float* out_boxes = out_classes + C_N * K_N;           // C*K*4

  retina_topk_decode<<<C_N, NT2, 0, stream>>>(cls, reg, anc, tk_score, tk_box);
  retina_nms<<<C_N, NT3, 0, stream>>>(tk_score, tk_box, nms, out_scores,
                                      out_classes, out_boxes);
}